// QNN_0_Prod_7602092113981
// MI455X (gfx1250) — compile-verified
//
#include <hip/hip_runtime.h>
#include <hip/hip_bf16.h>

typedef float v2f __attribute__((ext_vector_type(2)));
typedef float v8f __attribute__((ext_vector_type(8)));

// One wave = 16 unique samples (lanes 16-31 mirror lanes 0-15 so that the
// WMMA B-operand K=2 row (t2) is locally available with no cross-lane moves).
// Block = 256 threads = 8 waves = 128 samples.
__global__ __launch_bounds__(256) void qnn_prod_kernel(
    const float* __restrict__ batch,    // [B,3]
    const float* __restrict__ weights,  // [16,6] (cols 0..2 used)
    float* __restrict__ out,            // [B,3]
    int nSamples)
{
    __shared__ float sV[16 * 12];  // 16 layers, 3 rows padded to 4 floats

    const int tid = threadIdx.x;

    // ---- Build the 16 shared layer rotations V_l = R12(c)R02(b)R01(a) ----
    if (tid < 16) {
        const float a = weights[tid * 6 + 0];
        const float b = weights[tid * 6 + 1];
        const float c = weights[tid * 6 + 2];
        float sa, ca, sb, cb, sc, cc;
        __sincosf(a, &sa, &ca);
        __sincosf(b, &sb, &cb);
        __sincosf(c, &sc, &cc);
        float* V = &sV[tid * 12];
        V[0] = cb * ca;                    V[1] = cb * sa;                    V[2]  = sb;
        V[4] = -cc * sa - sc * sb * ca;    V[5] = cc * ca - sc * sb * sa;     V[6]  = sc * cb;
        V[8] =  sc * sa - cc * sb * ca;    V[9] = -sc * ca - cc * sb * sa;    V[10] = cc * cb;
        V[3] = 0.0f; V[7] = 0.0f; V[11] = 0.0f;
    }
    __syncthreads();

    const int lane = tid & 31;
    const int wave = tid >> 5;
    const int m    = lane & 15;        // A-matrix row index (and sample slot)
    const bool hi  = lane >= 16;       // hi half-wave: supplies K=2,3 of A/B
    const int mm   = m & 7;            // V replicated into rows 0-2 and 8-10
    const bool valid = (mm < 3);

    // ---- Hoist per-lane A-operand VGPR pairs for all 16 layers ----
    // A(16x4) layout: lanes 0-15: {K0,K1}; lanes 16-31: {K2,K3}; row M = lane&15.
    v2f A[16];
#pragma unroll
    for (int l = 0; l < 16; ++l) {
        const float* V = &sV[l * 12];
        float ax = valid ? V[mm * 4 + (hi ? 2 : 0)] : 0.0f;
        float ay = (valid && !hi) ? V[mm * 4 + 1] : 0.0f;
        A[l].x = ax;
        A[l].y = ay;
    }

    // ---- Per-sample encoding rotation U = R12(x2)R02(x1)R01(x0) ----
    int sample = blockIdx.x * 128 + wave * 16 + m;
    int sclamp = sample < nSamples ? sample : (nSamples - 1);
    const float* xp = batch + (size_t)sclamp * 3;
    const float x0 = xp[0], x1 = xp[1], x2 = xp[2];
    float sa, ca, sb, cb, sc, cc;
    __sincosf(x0, &sa, &ca);
    __sincosf(x1, &sb, &cb);
    __sincosf(x2, &sc, &cc);
    const float u00 = cb * ca,                 u01 = cb * sa,                 u02 = sb;
    const float u10 = -cc * sa - sc * sb * ca, u11 = cc * ca - sc * sb * sa,  u12 = sc * cb;
    const float u20 =  sc * sa - cc * sb * ca, u21 = -sc * ca - cc * sb * sa, u22 = cc * cb;

    // Per-lane B-row coefficients: each lane only produces ITS two B values.
    //   lo lanes: B.x = (U row0)·s = t0 ; B.y = (U row1)·s = t1   (K=0,1)
    //   hi lanes: B.x = (U row2)·s = t2 ; B.y = 0                 (K=2,3; A's K=3 col is 0)
    const float ra0 = hi ? u20 : u00, ra1 = hi ? u21 : u01, ra2 = hi ? u22 : u02;
    const float rb0 = hi ? 0.0f : u10, rb1 = hi ? 0.0f : u11, rb2 = hi ? 0.0f : u12;

    // state = e0 (uniform start; layer 0 then computes B from U·e0 implicitly)
    float s0 = 1.0f, s1 = 0.0f, s2 = 0.0f;

#pragma unroll
    for (int l = 0; l < 16; ++l) {
        v2f Bv;
        Bv.x = fmaf(ra0, s0, fmaf(ra1, s1, ra2 * s2));
        Bv.y = fmaf(rb0, s0, fmaf(rb1, s1, rb2 * s2));
        v8f cz = {};
        // D = A(V_l, rows 0-2 & 8-10) x B(t columns): new states land in
        // VGPRs 0-2 of BOTH half-waves (rows 0-2 for lo, rows 8-10 for hi).
        v8f d = __builtin_amdgcn_wmma_f32_16x16x4_f32(
            false, A[l], false, Bv, (short)0, cz, false, false);
        s0 = d[0];
        s1 = d[1];
        s2 = d[2];
    }

    // ---- fid_k = (sqrt(max(s_k^2, 1e-10)) + 2e-5)^2, normalized ----
    const float r0 = __builtin_sqrtf(fmaxf(s0 * s0, 1e-10f)) + 2.0e-5f;
    const float r1 = __builtin_sqrtf(fmaxf(s1 * s1, 1e-10f)) + 2.0e-5f;
    const float r2 = __builtin_sqrtf(fmaxf(s2 * s2, 1e-10f)) + 2.0e-5f;
    const float f0 = r0 * r0, f1 = r1 * r1, f2 = r2 * r2;
    const float inv = 1.0f / (f0 + f1 + f2);

    if (!hi && sample < nSamples) {
        float* o = out + (size_t)sample * 3;
        o[0] = f0 * inv;
        o[1] = f1 * inv;
        o[2] = f2 * inv;
    }
}

extern "C" void kernel_launch(void* const* d_in, const int* in_sizes, int n_in,
                              void* d_out, int out_size, void* d_ws, size_t ws_size,
                              hipStream_t stream) {
    const float* batch   = (const float*)d_in[0];  // [B,3] float32
    const float* weights = (const float*)d_in[1];  // [16,6] float32
    float* out = (float*)d_out;                    // [B,3] float32

    const int nSamples = in_sizes[0] / 3;          // B = 524288
    const int samplesPerBlock = 128;               // 8 waves x 16 samples
    const int grid = (nSamples + samplesPerBlock - 1) / samplesPerBlock;

    qnn_prod_kernel<<<grid, 256, 0, stream>>>(batch, weights, out, nSamples);
}